// SEBlock_62569083568766
// MI455X (gfx1250) — compile-verified
//
#include <hip/hip_runtime.h>

typedef float v2f __attribute__((ext_vector_type(2)));
typedef float v4f __attribute__((ext_vector_type(4)));
typedef float v8f __attribute__((ext_vector_type(8)));

#define B_   32
#define C_   512
#define HID_ 32
#define HW_  4096   // 64*64

// ---------------------------------------------------------------------------
// Pass 1: global average pool. One 256-thread block per (b,c) plane.
// 4096 floats/plane = 16 floats/thread as 4x float4, coalesced.
// ---------------------------------------------------------------------------
__global__ __launch_bounds__(256) void se_pool(const float* __restrict__ x,
                                               float* __restrict__ v) {
  const int plane = blockIdx.x;                      // b*C_ + c
  const v4f* xp = (const v4f*)(x + (size_t)plane * HW_);
  const int t = threadIdx.x;
  float s = 0.f;
#pragma unroll
  for (int i = 0; i < 4; ++i) {
    v4f q = xp[t + i * 256];
    s += (q.x + q.y) + (q.z + q.w);
  }
  // wave32 reduction
#pragma unroll
  for (int off = 16; off > 0; off >>= 1)
    s += __shfl_down(s, off, 32);
  __shared__ float part[8];
  const int wv = t >> 5, lane = t & 31;
  if (lane == 0) part[wv] = s;
  __syncthreads();
  if (t == 0) {
    float tot = 0.f;
#pragma unroll
    for (int w = 0; w < 8; ++w) tot += part[w];
    v[plane] = tot * (1.0f / (float)HW_);
  }
}

// ---------------------------------------------------------------------------
// Pass 2: squeeze-excite MLP with fp32 WMMA (V_WMMA_F32_16X16X4_F32).
//   h = relu(v @ w1^T)   : (32x512)@(512x32)
//   g = sigmoid(h @ w2^T): (32x32)@(32x512)
// Single workgroup, 8 waves. GEMM1 splits K over waves (LDS reduce);
// GEMM2 splits the 32 N-tiles over waves.
//
// Fragment layouts (wave32, 32-bit elements, per ISA 7.12.2):
//   A 16x4 : lane&15 = M row; VGPR0 = K (0 | 2 for lanes>=16), VGPR1 = K+1
//   B 4x16 : lane&15 = N col; VGPR0 = K (0 | 2 for lanes>=16), VGPR1 = K+1
//   C/D    : lane&15 = N col; VGPR r = row M = r + 8*(lane>=16)
// ---------------------------------------------------------------------------
__global__ __launch_bounds__(256) void se_mlp(const float* __restrict__ v,
                                              const float* __restrict__ w1,
                                              const float* __restrict__ w2,
                                              float* __restrict__ g) {
  __shared__ float hpart[8 * B_ * HID_];   // per-wave partial h (32 KB)
  __shared__ float hfin[B_ * HID_];        // relu'd hidden     (4 KB)

  const int t     = threadIdx.x;
  const int wv    = t >> 5;
  const int lane  = t & 31;
  const int lhalf = lane >> 4;             // 0: lanes 0-15, 1: lanes 16-31
  const int l15   = lane & 15;

  // ---- GEMM1: h = v @ w1^T, each wave owns a K-chunk of 64 ----
  v8f acc[2][2];
#pragma unroll
  for (int mi = 0; mi < 2; ++mi)
#pragma unroll
    for (int nj = 0; nj < 2; ++nj)
#pragma unroll
      for (int r = 0; r < 8; ++r) acc[mi][nj][r] = 0.f;

  const int kchunk = wv * 64;
  for (int kk = 0; kk < 64; kk += 4) {
    const int kb = kchunk + kk + lhalf * 2;
    v2f a[2], b[2];
#pragma unroll
    for (int mi = 0; mi < 2; ++mi) {
      const int row = mi * 16 + l15;
      a[mi].x = v[row * C_ + kb];
      a[mi].y = v[row * C_ + kb + 1];
    }
#pragma unroll
    for (int nj = 0; nj < 2; ++nj) {
      const int col = nj * 16 + l15;           // hidden index
      b[nj].x = w1[col * C_ + kb];             // B[k][n] = w1[n][k]
      b[nj].y = w1[col * C_ + kb + 1];
    }
#pragma unroll
    for (int mi = 0; mi < 2; ++mi)
#pragma unroll
      for (int nj = 0; nj < 2; ++nj)
        acc[mi][nj] = __builtin_amdgcn_wmma_f32_16x16x4_f32(
            false, a[mi], false, b[nj], (short)0, acc[mi][nj], false, false);
  }

  // spill this wave's partial accumulators to LDS in (m,n) order
#pragma unroll
  for (int mi = 0; mi < 2; ++mi)
#pragma unroll
    for (int nj = 0; nj < 2; ++nj)
#pragma unroll
      for (int r = 0; r < 8; ++r) {
        const int m = mi * 16 + r + lhalf * 8;
        const int n = nj * 16 + l15;
        hpart[wv * (B_ * HID_) + m * HID_ + n] = acc[mi][nj][r];
      }
  __syncthreads();

  // cross-wave reduce + ReLU
  for (int i = t; i < B_ * HID_; i += 256) {
    float s = 0.f;
#pragma unroll
    for (int w = 0; w < 8; ++w) s += hpart[w * (B_ * HID_) + i];
    hfin[i] = fmaxf(s, 0.f);
  }
  __syncthreads();

  // ---- GEMM2: g = h @ w2^T, 32 N-tiles split over 8 waves ----
  for (int tile = wv; tile < C_ / 16; tile += 8) {
    v8f acc2[2];
#pragma unroll
    for (int mi = 0; mi < 2; ++mi)
#pragma unroll
      for (int r = 0; r < 8; ++r) acc2[mi][r] = 0.f;

    const int col = tile * 16 + l15;           // channel index
#pragma unroll
    for (int k0 = 0; k0 < HID_; k0 += 4) {
      const int kb = k0 + lhalf * 2;
      v2f a[2], b;
#pragma unroll
      for (int mi = 0; mi < 2; ++mi) {
        const int row = mi * 16 + l15;
        a[mi].x = hfin[row * HID_ + kb];
        a[mi].y = hfin[row * HID_ + kb + 1];
      }
      b.x = w2[col * HID_ + kb];               // B[k][n] = w2[n][k]
      b.y = w2[col * HID_ + kb + 1];
#pragma unroll
      for (int mi = 0; mi < 2; ++mi)
        acc2[mi] = __builtin_amdgcn_wmma_f32_16x16x4_f32(
            false, a[mi], false, b, (short)0, acc2[mi], false, false);
    }
    // sigmoid + store straight from C/D layout
#pragma unroll
    for (int mi = 0; mi < 2; ++mi)
#pragma unroll
      for (int r = 0; r < 8; ++r) {
        const int m = mi * 16 + r + lhalf * 8;
        const float val = acc2[mi][r];
        g[m * C_ + col] = 1.0f / (1.0f + __expf(-val));
      }
  }
}

// ---------------------------------------------------------------------------
// Pass 3: channel scale. One block per (b,c) plane; gate is block-uniform.
// Streaming data -> nontemporal hints to avoid thrashing L2.
// ---------------------------------------------------------------------------
__global__ __launch_bounds__(256) void se_scale(const float* __restrict__ x,
                                                const float* __restrict__ gate,
                                                float* __restrict__ out) {
  const int plane = blockIdx.x;
  const float gv = gate[plane];
  const v4f* xp = (const v4f*)(x + (size_t)plane * HW_);
  v4f* op = (v4f*)(out + (size_t)plane * HW_);
  const int t = threadIdx.x;
#pragma unroll
  for (int i = 0; i < 4; ++i) {
    v4f q = __builtin_nontemporal_load(&xp[t + i * 256]);
    q.x *= gv; q.y *= gv; q.z *= gv; q.w *= gv;
    __builtin_nontemporal_store(q, &op[t + i * 256]);
  }
}

// ---------------------------------------------------------------------------
extern "C" void kernel_launch(void* const* d_in, const int* in_sizes, int n_in,
                              void* d_out, int out_size, void* d_ws, size_t ws_size,
                              hipStream_t stream) {
  const float* x  = (const float*)d_in[0];   // (32, 512, 64, 64)
  const float* w1 = (const float*)d_in[1];   // (32, 512)
  const float* w2 = (const float*)d_in[2];   // (512, 32)
  float* out = (float*)d_out;

  float* v = (float*)d_ws;                   // 16384 floats: pooled means (B,C)
  float* g = v + (size_t)B_ * C_;            // 16384 floats: gates (B,C)

  se_pool <<<B_ * C_, 256, 0, stream>>>(x, v);
  se_mlp  <<<1,       256, 0, stream>>>(v, w1, w2, g);
  se_scale<<<B_ * C_, 256, 0, stream>>>(x, g, out);
}